// MLPAgent_71683004170935
// MI455X (gfx1250) — compile-verified
//
#include <hip/hip_runtime.h>
#include <hip/hip_bf16.h>

// ---------------- problem constants (from reference) ----------------
#define NUM_ENT   100000
#define ENT_ROWS  (NUM_ENT + 1)      // 100001
#define ENT_FEAT  96                 // ENT_DIM - TIME_DIM
#define KCHUNKS   (ENT_FEAT / 4)     // 24 WMMA k-steps
#define TIME_DIM  32
#define REL_DIM   128
#define REL_ROWS  401                // 2*NUM_REL + 1
#define BATCH     2048
#define NACT      200
#define DT_SPAN   729                // dt in [-364, 364]
#define DT_OFF    364
#define NEG_FILL  (-1e10f)

typedef __attribute__((ext_vector_type(2))) float v2f;
typedef __attribute__((ext_vector_type(8))) float v8f;

// =====================================================================
// Phase 1a: ent_dot{1,2}[e] = <ent_table[e], w[128:224]> , <ent_table[e], w[256:352]>
// Streaming GEMV [100001 x 96] x [96 x 2] via V_WMMA_F32_16X16X4_F32.
// One wave per 16-row tile. B operand is (kc, lane)-only -> prebuilt in LDS
// once per block; inner loop = global_load_b64(A) + ds_load_b64(B) + wmma.
// =====================================================================
__global__ void __launch_bounds__(256)
ent_dot_wmma_kernel(const float* __restrict__ ent_table,
                    const float* __restrict__ fc_w,
                    float* __restrict__ ent_dot1,
                    float* __restrict__ ent_dot2,
                    int nrows)
{
    // LDS image of the B operand: [kc][lane*2 + comp] -> conflict-free b64 reads
    __shared__ float sB[KCHUNKS * 64];

    // Cooperative, branchless fill (B: col 0 = w[128:224], col 1 = w[256:352], rest 0)
    for (int idx = threadIdx.x; idx < KCHUNKS * 64; idx += 256) {
        const int kc    = idx >> 6;
        const int rem   = idx & 63;
        const int lane  = rem >> 1;
        const int comp  = rem & 1;
        const int khalf = lane >> 4;
        const int ncol  = lane & 15;
        const int k     = kc * 4 + khalf * 2 + comp;
        const float w1  = fc_w[128 + k];
        const float w2  = fc_w[256 + k];
        sB[idx] = (ncol == 0) ? w1 : ((ncol == 1) ? w2 : 0.0f);
    }
    __syncthreads();

    const int lane  = threadIdx.x & 31;
    const int waveI = threadIdx.x >> 5;
    const int wavesPerBlock = blockDim.x >> 5;
    const int ntiles = (nrows + 15) >> 4;

    int tile = blockIdx.x * wavesPerBlock + waveI;
    if (tile >= ntiles) tile = ntiles - 1;    // clamp: dup compute, masked stores
    const int row_base = tile * 16;

    const int m     = lane & 15;              // A: M index
    const int khalf = lane >> 4;              // A: lanes 0-15 hold K{0,1}, 16-31 hold K{2,3}

    int row = row_base + m;
    if (row >= nrows) row = nrows - 1;        // clamp tail row
    // Per-lane A base: row start + 2*khalf floats; k-chunk kc at byte offset kc*16
    const float* __restrict__ abase = ent_table + (size_t)row * ENT_FEAT + 2 * khalf;
    const v2f*   __restrict__ bbase = ((const v2f*)sB) + lane;   // + kc*32

    v8f acc = {};
    #pragma unroll
    for (int kc = 0; kc < KCHUNKS; ++kc) {
        const v2f a  = *(const v2f*)(abase + kc * 4);   // global_load_b64, imm offset
        const v2f bv = bbase[kc * 32];                  // ds_load_b64, conflict-free
        acc = __builtin_amdgcn_wmma_f32_16x16x4_f32(
            /*neg_a=*/false, a, /*neg_b=*/false, bv,
            /*c_mod=*/(short)0, acc, /*reuse_a=*/false, /*reuse_b=*/false);
    }

    // D layout: VGPR v: lanes 0-15 -> M=v, N=lane; lanes 16-31 -> M=v+8, N=lane-16
    const int ncol = m;
    if (ncol < 2) {
        float* __restrict__ outp = (ncol == 0) ? ent_dot1 : ent_dot2;
        #pragma unroll
        for (int v = 0; v < 8; ++v) {
            const int mm = row_base + khalf * 8 + v;
            if (mm < nrows) outp[mm] = acc[v];
        }
    }
}

// =====================================================================
// Phase 1b: rel_dot1[r] = <rel_table[r], w[0:128]> ; rel_dot2[r] = <.., w[384:512]>
// One 128-thread block per relation row (401 rows, ~205 KB total: trivial).
// =====================================================================
__global__ void __launch_bounds__(128)
rel_dot_kernel(const float* __restrict__ rel_table,
               const float* __restrict__ fc_w,
               float* __restrict__ rel_dot1,
               float* __restrict__ rel_dot2)
{
    __shared__ float s1[128];
    __shared__ float s2[128];
    const int r = blockIdx.x;
    const int t = threadIdx.x;
    const float x = rel_table[r * REL_DIM + t];
    s1[t] = x * fc_w[t];
    s2[t] = x * fc_w[384 + t];
    __syncthreads();
    for (int s = 64; s > 0; s >>= 1) {
        if (t < s) { s1[t] += s1[t + s]; s2[t] += s2[t + s]; }
        __syncthreads();
    }
    if (t == 0) { rel_dot1[r] = s1[0]; rel_dot2[r] = s2[0]; }
}

// =====================================================================
// Phase 1c: time_dot{1,2}[dt+364] = sum_j cos(dt*tw[j]+tb[j]) * w_time{nb,cur}[j]
// dt has only 729 possible integer values -> lookup table.
// =====================================================================
__global__ void __launch_bounds__(256)
time_dot_kernel(const float* __restrict__ time_w,
                const float* __restrict__ time_b,
                const float* __restrict__ fc_w,
                float* __restrict__ time_dot1,
                float* __restrict__ time_dot2)
{
    const int i = blockIdx.x * blockDim.x + threadIdx.x;
    if (i >= DT_SPAN) return;
    const float dt = (float)(i - DT_OFF);
    float d1 = 0.0f, d2 = 0.0f;
    #pragma unroll
    for (int j = 0; j < TIME_DIM; ++j) {
        const float c = __cosf(dt * time_w[j] + time_b[j]);
        d1 += c * fc_w[224 + j];   // neighbor time slice  w[224:256]
        d2 += c * fc_w[352 + j];   // current  time slice  w[352:384]
    }
    time_dot1[i] = d1;
    time_dot2[i] = d2;
}

// =====================================================================
// Phase 2: per-batch-row scores via table lookups, softmax, Gumbel-max
// categorical sample (deterministic hash RNG), loss. One block per b.
// =====================================================================
__device__ __forceinline__ float hash_u01(unsigned long long x)
{
    x ^= x >> 33; x *= 0xff51afd7ed558ccdULL;
    x ^= x >> 33; x *= 0xc4ceb9fe1a85ec53ULL;
    x ^= x >> 33;
    return ((float)((x >> 40) & 0xFFFFFFULL) + 0.5f) * (1.0f / 16777216.0f);
}

__global__ void __launch_bounds__(256)
score_kernel(const int*   __restrict__ prev_relation,
             const int*   __restrict__ current_entities,
             const int*   __restrict__ current_timestamps,
             const int*   __restrict__ query_timestamps,
             const int*   __restrict__ neibor_entitys,
             const int*   __restrict__ neibor_timestamps,
             const int*   __restrict__ neibor_relations,
             const float* __restrict__ ent_dot1,
             const float* __restrict__ ent_dot2,
             const float* __restrict__ rel_dot1,
             const float* __restrict__ rel_dot2,
             const float* __restrict__ time_dot1,
             const float* __restrict__ time_dot2,
             const float* __restrict__ fc_b,
             float* __restrict__ loss_out,
             float* __restrict__ logits_out,
             float* __restrict__ action_out)
{
    __shared__ float red[256];
    __shared__ int   redi[256];
    __shared__ float slog[256];
    __shared__ float s_state;

    const int b = blockIdx.x;
    const int t = threadIdx.x;
    const int qts = query_timestamps[b];

    if (t == 0) {
        const int dtc = qts - current_timestamps[b] + DT_OFF;
        s_state = ent_dot2[current_entities[b]] + time_dot2[dtc]
                + rel_dot2[prev_relation[b]] + fc_b[0];
    }
    __syncthreads();

    float score = NEG_FILL;
    if (t < NACT) {
        const int idx = b * NACT + t;
        const int ne  = neibor_entitys[idx];
        const int dtn = qts - neibor_timestamps[idx] + DT_OFF;
        const int nr  = neibor_relations[idx];
        const float pre = s_state + ent_dot1[ne] + time_dot1[dtn] + rel_dot1[nr];
        const float sig = 1.0f / (1.0f + __expf(-pre));
        score = (ne == NUM_ENT) ? NEG_FILL : sig;
    }

    // row max
    red[t] = score; __syncthreads();
    for (int s = 128; s > 0; s >>= 1) {
        if (t < s) red[t] = fmaxf(red[t], red[t + s]);
        __syncthreads();
    }
    const float mx = red[0]; __syncthreads();

    // sum of exp
    red[t] = (t < NACT) ? __expf(score - mx) : 0.0f; __syncthreads();
    for (int s = 128; s > 0; s >>= 1) {
        if (t < s) red[t] += red[t + s];
        __syncthreads();
    }
    const float lse = __logf(red[0]); __syncthreads();

    const float logit = score - mx - lse;
    slog[t] = logit;
    if (t < NACT) logits_out[b * NACT + t] = logit;

    // Gumbel-max categorical sample (deterministic per (b, t))
    float pert = -3.0e38f;
    if (t < NACT) {
        const unsigned long long key =
            0x2aULL * 0x9E3779B97F4A7C15ULL + (unsigned long long)(b * NACT + t);
        const float u = hash_u01(key);
        pert = score - __logf(-__logf(u));
    }
    red[t] = pert; redi[t] = t; __syncthreads();
    for (int s = 128; s > 0; s >>= 1) {
        if (t < s && red[t + s] > red[t]) { red[t] = red[t + s]; redi[t] = redi[t + s]; }
        __syncthreads();
    }

    if (t == 0) {
        const int aid = redi[0];
        action_out[b] = (float)aid;
        loss_out[b]   = -slog[aid];
    }
}

// =====================================================================
// launcher
// =====================================================================
extern "C" void kernel_launch(void* const* d_in, const int* in_sizes, int n_in,
                              void* d_out, int out_size, void* d_ws, size_t ws_size,
                              hipStream_t stream)
{
    (void)in_sizes; (void)n_in; (void)out_size; (void)ws_size;

    const int*   prev_relation      = (const int*)  d_in[0];
    const int*   current_entities   = (const int*)  d_in[1];
    const int*   current_timestamps = (const int*)  d_in[2];
    // d_in[3] query_relation, d_in[4] query_entity: unused by the reference math
    const int*   query_timestamps   = (const int*)  d_in[5];
    const int*   neibor_entitys     = (const int*)  d_in[6];
    const int*   neibor_timestamps  = (const int*)  d_in[7];
    const int*   neibor_relations   = (const int*)  d_in[8];
    const float* ent_table          = (const float*)d_in[9];
    const float* time_w             = (const float*)d_in[10];
    const float* time_b             = (const float*)d_in[11];
    const float* rel_table          = (const float*)d_in[12];
    const float* fc_w               = (const float*)d_in[13];
    const float* fc_b               = (const float*)d_in[14];

    // workspace layout (floats)
    float* ws        = (float*)d_ws;
    float* ent_dot1  = ws;                       // 100001
    float* ent_dot2  = ent_dot1 + ENT_ROWS;      // 100001
    float* rel_dot1  = ent_dot2 + ENT_ROWS;      // 401
    float* rel_dot2  = rel_dot1 + REL_ROWS;      // 401
    float* time_dot1 = rel_dot2 + REL_ROWS;      // 729
    float* time_dot2 = time_dot1 + DT_SPAN;      // 729

    // output layout: loss[B] | logits[B*NACT] | action_id[B]
    float* loss_out   = (float*)d_out;
    float* logits_out = loss_out + BATCH;
    float* action_out = logits_out + (size_t)BATCH * NACT;

    // Phase 1a: WMMA GEMV over ent_table (one wave per 16-row tile, 8 waves/block)
    {
        const int tiles = (ENT_ROWS + 15) / 16;
        const int wavesPerBlock = 8;
        const int grid = (tiles + wavesPerBlock - 1) / wavesPerBlock;
        ent_dot_wmma_kernel<<<grid, 256, 0, stream>>>(
            ent_table, fc_w, ent_dot1, ent_dot2, ENT_ROWS);
    }
    // Phase 1b: relation-row dots
    rel_dot_kernel<<<REL_ROWS, 128, 0, stream>>>(rel_table, fc_w, rel_dot1, rel_dot2);
    // Phase 1c: time lookup table
    time_dot_kernel<<<(DT_SPAN + 255) / 256, 256, 0, stream>>>(
        time_w, time_b, fc_w, time_dot1, time_dot2);
    // Phase 2: scores + softmax + sample + loss
    score_kernel<<<BATCH, 256, 0, stream>>>(
        prev_relation, current_entities, current_timestamps, query_timestamps,
        neibor_entitys, neibor_timestamps, neibor_relations,
        ent_dot1, ent_dot2, rel_dot1, rel_dot2, time_dot1, time_dot2, fc_b,
        loss_out, logits_out, action_out);
}